// LengthRegulator_10256381903530
// MI455X (gfx1250) — compile-verified
//
#include <hip/hip_runtime.h>

// LengthRegulator for MI455X (gfx1250, wave32).
// x: [B,S,D] f32, durations: [B,S] i32  ->  out: [B,F,D] f32 (+ out_lens[B] as f32)
// Pure gather/stream op: no matrix FLOPs -> no WMMA; the target is the HBM path.
//   - wave32 per output frame; 3 x B128 load + 3 x B128 NT store per lane
//   - NT stores: 177MB write-once stream must not evict x (50MB, ~3.5x reuse)
//     from the 192MB L2
//   - wave-uniform control (readfirstlane) -> binary search on SALU/SMEM path
//   - shuffle-based wave32 scan for the duration cumsum (2 barriers total)

typedef float v4f __attribute__((ext_vector_type(4)));

#define LR_B 32
#define LR_S 1024
#define LR_D 384

// ---------------------------------------------------------------------------
// Kernel 1: per-batch inclusive scan of durations -> cum[B,S], out_lens[B]
// One 1024-thread block (32 wave32s) per batch.
// Wave-level scan in registers (shfl_up), cross-wave via a 32-entry LDS bounce.
// ---------------------------------------------------------------------------
__global__ __launch_bounds__(LR_S) void lr_scan_kernel(
    const int* __restrict__ dur,
    int* __restrict__ cum,
    float* __restrict__ out_lens_f)
{
    __shared__ int wsum[32];
    const int b    = blockIdx.x;
    const int s    = threadIdx.x;
    const int lane = s & 31;
    const int w    = s >> 5;

    int v = dur[b * LR_S + s];

    // inclusive scan within the wave32 (5 shuffle steps, no LDS)
#pragma unroll
    for (int off = 1; off < 32; off <<= 1) {
        int t = __shfl_up(v, off, 32);
        if (lane >= off) v += t;
    }

    if (lane == 31) wsum[w] = v;       // per-wave total
    __syncthreads();

    if (w == 0) {                      // wave 0 scans the 32 wave totals
        int t = wsum[lane];
#pragma unroll
        for (int off = 1; off < 32; off <<= 1) {
            int u = __shfl_up(t, off, 32);
            if (lane >= off) t += u;
        }
        wsum[lane] = t;
    }
    __syncthreads();

    const int base = (w > 0) ? wsum[w - 1] : 0;
    const int c = v + base;
    cum[b * LR_S + s] = c;
    if (s == LR_S - 1) {
        out_lens_f[b] = (float)c;      // out_lens concatenated after out, as f32
    }
}

// ---------------------------------------------------------------------------
// Kernel 2: one wave32 per output frame.
//   idx = searchsorted(cum[b,:], p, side='right')  (10-step scalar bsearch)
//   copy x[b, idx, :] -> out[b, p, :]  (3 x b128 per lane, NT stores)
//   frames beyond out_len[b] are zero-filled (scalar branch, no divergence).
// ---------------------------------------------------------------------------
__global__ __launch_bounds__(256) void lr_gather_kernel(
    const float* __restrict__ x,
    const int* __restrict__ cum,
    float* __restrict__ out,
    int F, int nframes)
{
    const int lane = threadIdx.x & 31;
    // Pin the wave id as wave-uniform so b/p/total and the binary search
    // become SALU + SMEM (s_load) work instead of redundant per-lane VALU/VMEM.
    const int wv   = __builtin_amdgcn_readfirstlane((int)(threadIdx.x >> 5));
    const int wave = blockIdx.x * (256 / 32) + wv;
    if (wave >= nframes) return;       // uniform -> s_cbranch

    const int b = wave / F;            // uniform
    const int p = wave - b * F;        // uniform

    const int* __restrict__ cb = cum + b * LR_S;
    const int total = cb[LR_S - 1];    // uniform load

    float* __restrict__ dst = out + ((size_t)b * F + p) * LR_D;

    if (p >= total) {                  // uniform branch: padding frame
        v4f z = {0.f, 0.f, 0.f, 0.f};
#pragma unroll
        for (int k = 0; k < 3; ++k)
            __builtin_nontemporal_store(z, (v4f*)(dst + k * 128 + lane * 4));
        return;
    }

    // searchsorted right: smallest t with cum[t] > p. 1024 elems -> 10 steps.
    int lo = 0, hi = LR_S;
#pragma unroll
    for (int it = 0; it < 10; ++it) {
        int mid = (lo + hi) >> 1;
        if (cb[mid] <= p) lo = mid + 1; else hi = mid;   // uniform cmp/select
    }
    const int idx = (lo < LR_S - 1) ? lo : (LR_S - 1);

    const float* __restrict__ src = x + ((size_t)b * LR_S + idx) * LR_D;

    // 384 floats / 32 lanes = 12 floats = 3 x float4 per lane.
    // Loads: regular temporal (x reused ~3.5x, resident in 192MB L2).
    // Stores: non-temporal streaming.
#pragma unroll
    for (int k = 0; k < 3; ++k) {
        v4f v = *(const v4f*)(src + k * 128 + lane * 4);
        __builtin_nontemporal_store(v, (v4f*)(dst + k * 128 + lane * 4));
    }
}

// ---------------------------------------------------------------------------
extern "C" void kernel_launch(void* const* d_in, const int* in_sizes, int n_in,
                              void* d_out, int out_size, void* d_ws, size_t ws_size,
                              hipStream_t stream)
{
    const float* x   = (const float*)d_in[0];   // [B,S,D] f32
    const int*   dur = (const int*)d_in[1];     // [B,S]   i32
    float*       out = (float*)d_out;           // [B,F,D] f32 ++ [B] lens (f32)

    // F is data-dependent; recover it from out_size = B*F*D + B.
    const int F = (out_size - LR_B) / (LR_B * LR_D);

    int* cum = (int*)d_ws;                      // B*S ints = 128 KB scratch
    float* out_lens_f = out + (size_t)LR_B * F * LR_D;

    lr_scan_kernel<<<LR_B, LR_S, 0, stream>>>(dur, cum, out_lens_f);

    const int nframes = LR_B * F;
    const int waves_per_block = 256 / 32;
    const int grid = (nframes + waves_per_block - 1) / waves_per_block;
    lr_gather_kernel<<<grid, 256, 0, stream>>>(x, cum, out, F, nframes);
}